// GraphSAGEEmb_60421599920554
// MI455X (gfx1250) — compile-verified
//
#include <hip/hip_runtime.h>
#include <hip/hip_bf16.h>
#include <math.h>

// GraphSAGE (3x SAGEConv max-aggr, D=128) for MI455X / gfx1250.
// Scatter-max via monotone u32-encoded global_atomic_max_u32 (L2-resident),
// GEMMs via V_WMMA_F32_16X16X4_F32 with 5 row-tiles per wave for matrix-pipe
// ILP (5 independent accumulators) and 5x weight-traffic amortization.

#define N_NODES 100000
#define DIM     128
#define NE      1600000
#define LDS_STRIDE 132   // 132 % 64 == 4 -> conflict-free fragment reads
#define ROWT    5        // row tiles per wave; 100000 = 80 * 1250 exactly
#define MROWS   (16 * ROWT)

typedef float v2f __attribute__((ext_vector_type(2)));
typedef float v8f __attribute__((ext_vector_type(8)));

// Order-preserving float -> u32 map (total order, -inf smallest).
__device__ __forceinline__ unsigned enc_f32(float f) {
  unsigned u = __float_as_uint(f);
  return (u & 0x80000000u) ? ~u : (u | 0x80000000u);
}
__device__ __forceinline__ float dec_f32(unsigned k) {
  unsigned u = (k & 0x80000000u) ? (k ^ 0x80000000u) : ~k;
  return __uint_as_float(u);
}

__device__ __forceinline__ unsigned mix32(unsigned v) {
  v ^= v >> 16; v *= 0x7feb352du;
  v ^= v >> 15; v *= 0x846ca68bu;
  v ^= v >> 16;
  return v;
}

// Fill agg buffer with encode(-inf) = ~bits(-inf) = 0x007FFFFF.
__global__ __launch_bounds__(256) void sage_init_agg(uint4* __restrict__ agg, int n4) {
  int i = blockIdx.x * 256 + threadIdx.x;
  if (i < n4) {
    const unsigned v = 0x007FFFFFu;
    agg[i] = make_uint4(v, v, v, v);
  }
}

// One wave32 per edge: gather 512B source row (float4/lane), 4 atomic-max/lane.
__global__ __launch_bounds__(256) void sage_scatter_max(
    const float* __restrict__ x, const int* __restrict__ ei,
    unsigned* __restrict__ agg) {
  const int e    = blockIdx.x * 8 + (threadIdx.x >> 5);
  const int lane = threadIdx.x & 31;
  const int src = ei[e];
  const int dst = ei[NE + e];
  const float4 v = *(const float4*)(x + (size_t)src * DIM + lane * 4);
  unsigned* p = agg + (size_t)dst * DIM + lane * 4;
  atomicMax(p + 0, enc_f32(v.x));
  atomicMax(p + 1, enc_f32(v.y));
  atomicMax(p + 2, enc_f32(v.z));
  atomicMax(p + 3, enc_f32(v.w));
}

// out[80 x 128 tile] = dec(agg) @ Wl + x @ Wr + b  (+ optional ELU/dropout).
// Block = 256 threads = 8 waves; wave w owns columns [16w, 16w+16) and all
// ROWT row tiles (independent accumulators -> WMMA ILP, B-fragment reuse).
__global__ __launch_bounds__(256) void sage_gemm(
    const unsigned* __restrict__ agg_enc, const float* __restrict__ xin,
    const float* __restrict__ Wl, const float* __restrict__ Wr,
    const float* __restrict__ bias, float* __restrict__ out,
    int applyAct, unsigned seed) {
  __shared__ float ldsA[2 * MROWS * LDS_STRIDE];
  float* A1 = ldsA;                         // decoded max-aggregate tile
  float* A2 = ldsA + MROWS * LDS_STRIDE;    // x tile

  const int tid   = threadIdx.x;
  const int nbase = blockIdx.x * MROWS;

  // Cooperative LDS fill: decode agg (non-finite -> 0, torch_scatter zero-init).
  for (int idx = tid; idx < MROWS * DIM; idx += 256) {
    const int r = idx >> 7, cc = idx & 127;
    const float f = dec_f32(agg_enc[(size_t)(nbase + r) * DIM + cc]);
    const unsigned fb = __float_as_uint(f);
    A1[r * LDS_STRIDE + cc] = ((fb & 0x7F800000u) == 0x7F800000u) ? 0.0f : f;
    A2[r * LDS_STRIDE + cc] = xin[(size_t)(nbase + r) * DIM + cc];
  }
  __syncthreads();

  const int lane = tid & 31;
  const int wv   = tid >> 5;
  const int m    = lane & 15;           // A-fragment row within tile
  const int koff = (lane >> 4) << 1;    // lanes 0-15: K={0,1}; 16-31: K={2,3}
  const int col  = wv * 16 + m;         // B-fragment / output column

  __builtin_prefetch(Wl + col, 0, 0);
  __builtin_prefetch(Wr + col, 0, 0);

  v8f acc[ROWT];
  #pragma unroll
  for (int t = 0; t < ROWT; ++t) acc[t] = (v8f){};

  #pragma unroll 4
  for (int k = 0; k < DIM; k += 4) {
    v2f b;
    b.x = Wl[(k + koff) * DIM + col];
    b.y = Wl[(k + koff + 1) * DIM + col];
    #pragma unroll
    for (int t = 0; t < ROWT; ++t) {
      v2f a;
      a.x = A1[(t * 16 + m) * LDS_STRIDE + k + koff];
      a.y = A1[(t * 16 + m) * LDS_STRIDE + k + koff + 1];
      acc[t] = __builtin_amdgcn_wmma_f32_16x16x4_f32(false, a, false, b,
                                                     (short)0, acc[t], false, false);
    }
  }
  #pragma unroll 4
  for (int k = 0; k < DIM; k += 4) {
    v2f b;
    b.x = Wr[(k + koff) * DIM + col];
    b.y = Wr[(k + koff + 1) * DIM + col];
    #pragma unroll
    for (int t = 0; t < ROWT; ++t) {
      v2f a;
      a.x = A2[(t * 16 + m) * LDS_STRIDE + k + koff];
      a.y = A2[(t * 16 + m) * LDS_STRIDE + k + koff + 1];
      acc[t] = __builtin_amdgcn_wmma_f32_16x16x4_f32(false, a, false, b,
                                                     (short)0, acc[t], false, false);
    }
  }

  const float bv   = bias[col];
  const int   rofs = (lane >> 4) << 3;  // C layout: lanes 16-31 hold M = r+8
  #pragma unroll
  for (int t = 0; t < ROWT; ++t) {
    #pragma unroll
    for (int r = 0; r < 8; ++r) {
      float v = acc[t][r] + bv;
      const int row = nbase + t * 16 + r + rofs;
      if (applyAct) {
        v = (v > 0.0f) ? v : expm1f(v);                       // ELU
        const unsigned h = mix32(((unsigned)row * DIM + col) ^ seed);
        v = ((h & 0xFFFFFFu) < 13421772u) ? v * 1.25f : 0.0f; // dropout p=0.2
      }
      out[(size_t)row * DIM + col] = v;
    }
  }
}

extern "C" void kernel_launch(void* const* d_in, const int* in_sizes, int n_in,
                              void* d_out, int out_size, void* d_ws, size_t ws_size,
                              hipStream_t stream) {
  (void)in_sizes; (void)n_in; (void)out_size; (void)ws_size;
  const float* feat = (const float*)d_in[0];
  const int*   ei   = (const int*)d_in[1];   // (2,E) int32 (JAX x64 disabled)
  const float* Wl0 = (const float*)d_in[2];
  const float* Wr0 = (const float*)d_in[3];
  const float* b0  = (const float*)d_in[4];
  const float* Wl1 = (const float*)d_in[5];
  const float* Wr1 = (const float*)d_in[6];
  const float* b1  = (const float*)d_in[7];
  const float* Wl2 = (const float*)d_in[8];
  const float* Wr2 = (const float*)d_in[9];
  const float* b2  = (const float*)d_in[10];

  const size_t mat = (size_t)N_NODES * DIM;
  unsigned* agg = (unsigned*)d_ws;                                 // 51.2 MB
  float*    h1  = (float*)((char*)d_ws + mat * sizeof(float));     // 51.2 MB
  float*    h2  = (float*)((char*)d_ws + 2 * mat * sizeof(float)); // 51.2 MB
  float*    y   = (float*)d_out;

  const int n4 = (int)(mat / 4);
  const dim3 blk(256);
  const dim3 gInit((n4 + 255) / 256);
  const dim3 gSct(NE / 8);
  const dim3 gGemm(N_NODES / MROWS);   // 1250 blocks, exact

  // Layer 0
  sage_init_agg<<<gInit, blk, 0, stream>>>((uint4*)agg, n4);
  sage_scatter_max<<<gSct, blk, 0, stream>>>(feat, ei, agg);
  sage_gemm<<<gGemm, blk, 0, stream>>>(agg, feat, Wl0, Wr0, b0, h1, 1, 0x9E3779B9u);
  // Layer 1
  sage_init_agg<<<gInit, blk, 0, stream>>>((uint4*)agg, n4);
  sage_scatter_max<<<gSct, blk, 0, stream>>>(h1, ei, agg);
  sage_gemm<<<gGemm, blk, 0, stream>>>(agg, h1, Wl1, Wr1, b1, h2, 1, 0x85EBCA6Bu);
  // Layer 2 (no activation/dropout)
  sage_init_agg<<<gInit, blk, 0, stream>>>((uint4*)agg, n4);
  sage_scatter_max<<<gSct, blk, 0, stream>>>(h2, ei, agg);
  sage_gemm<<<gGemm, blk, 0, stream>>>(agg, h2, Wl2, Wr2, b2, y, 0, 0u);
}